// NoiseConv_34943853920605
// MI455X (gfx1250) — compile-verified
//
#include <hip/hip_runtime.h>
#include <hip/hip_bf16.h>

typedef __attribute__((ext_vector_type(2))) float v2f;
typedef __attribute__((ext_vector_type(4))) float v4f;
typedef __attribute__((ext_vector_type(8))) float v8f;

// Problem dims (fixed by reference):
//   B=8, I=64, S=64, P=16, O=128, SP=S*P=1024, N=B*SP=8192
#define DIM_I 64
#define DIM_O 128
#define DIM_SP 1024
#define DIM_N 8192
#define NOISE_F 0.1f

// ---------------------------------------------------------------------------
// Kernel 1: conv_out[b,o,sp] = sum_i W[o,i] * x[b,i,sp] + bias[o]
// One wave per 16x16 (o x sp) tile, K=64 via 16 steps of V_WMMA_F32_16X16X4_F32.
// Tiles: 8 (b) * 8 (o-tiles) * 64 (sp-tiles) = 4096 waves.
// ---------------------------------------------------------------------------
__global__ __launch_bounds__(256)
void noiseconv_wmma_conv(const float* __restrict__ x,
                         const float* __restrict__ W,
                         const float* __restrict__ bias,
                         float* __restrict__ out) {
  const int wave = blockIdx.x * (blockDim.x >> 5) + (threadIdx.x >> 5);
  if (wave >= 8 * 8 * 64) return;
  const int lane = threadIdx.x & 31;

  const int ct = wave & 63;          // sp-tile 0..63
  const int ot = (wave >> 6) & 7;    // o-tile  0..7
  const int bb = wave >> 9;          // batch   0..7
  const int c0 = ct * 16;
  const int o0 = ot * 16;

  const float* xb = x + (size_t)bb * DIM_I * DIM_SP;   // [I, SP] row-major

  const int mn    = lane & 15;            // row (A) / col (B) within tile
  const int khalf = (lane >> 4) * 2;      // K sub-block per lane half

  v8f acc = {};
  #pragma unroll 4
  for (int k0 = 0; k0 < DIM_I; k0 += 4) {
    // A tile: W[o0+mn][k0+khalf .. +1]  (even float offset -> 8B aligned)
    v2f a = *(const v2f*)(W + (size_t)(o0 + mn) * DIM_I + k0 + khalf);
    // B tile: xb[k0+khalf .. +1][c0+mn] (stride DIM_SP between K rows)
    const float* xp = xb + (size_t)(k0 + khalf) * DIM_SP + c0 + mn;
    v2f b;
    b.x = xp[0];
    b.y = xp[DIM_SP];
    acc = __builtin_amdgcn_wmma_f32_16x16x4_f32(
        /*neg_a=*/false, a, /*neg_b=*/false, b,
        /*c_mod=*/(short)0, acc, /*reuse_a=*/false, /*reuse_b=*/false);
  }

  // C/D layout: VGPR r -> M = r + 8*(lane>=16), N = lane&15
  const int nl  = lane & 15;
  const int mhi = (lane >> 4) * 8;
  float* ob = out + (size_t)bb * DIM_O * DIM_SP + c0 + nl;
  #pragma unroll
  for (int r = 0; r < 8; ++r) {
    const int mrow = r + mhi;
    ob[(size_t)(o0 + mrow) * DIM_SP] = acc[r] + bias[o0 + mrow];
  }
}

// ---------------------------------------------------------------------------
// Kernel 2: out[b,o,sp] += 0.1 * sum_i W[o,i] * x[b,i,sp] * eps[n,o,i]
// One wave per n (n = b*SP + sp). Lanes split into two halves; each half
// covers one o-row of eps with a float4 NT load (512B coalesced per wave
// per iteration). x[n,:] held in registers across all 128 o's.
// ---------------------------------------------------------------------------
__global__ __launch_bounds__(256)
void noiseconv_noisy_stream(const float* __restrict__ x,
                            const float* __restrict__ W,
                            const float* __restrict__ eps,
                            float* __restrict__ out) {
  const int wid = blockIdx.x * (blockDim.x >> 5) + (threadIdx.x >> 5);
  if (wid >= DIM_N) return;
  const int lane = threadIdx.x & 31;

  const int nn = wid;                 // flattened n
  const int bb = nn >> 10;            // n / SP
  const int sp = nn & (DIM_SP - 1);   // n % SP

  const int il   = lane & 15;         // i-chunk index within half-wave
  const int ohalf = lane >> 4;        // which o of the pair this half covers
  const int ib   = 4 * il;            // i base: each lane owns i = ib..ib+3

  // x_flat[n, i] = x[bb, i, sp] ; preload this lane's 4 i-values (reused 128x)
  const float* xb = x + (size_t)bb * DIM_I * DIM_SP + sp;
  v4f xv;
  xv.x = xb[(size_t)(ib + 0) * DIM_SP];
  xv.y = xb[(size_t)(ib + 1) * DIM_SP];
  xv.z = xb[(size_t)(ib + 2) * DIM_SP];
  xv.w = xb[(size_t)(ib + 3) * DIM_SP];

  const float* ep = eps + (size_t)nn * DIM_O * DIM_I;  // this n's [O, I] slab
  float* ob = out + (size_t)bb * DIM_O * DIM_SP + sp;

  #pragma unroll 2
  for (int o0 = 0; o0 < DIM_O; o0 += 2) {
    const int o = o0 + ohalf;
    // W row chunk (hot in cache), eps chunk (streamed, non-temporal)
    v4f wv = *(const v4f*)(W + (size_t)o * DIM_I + ib);
    v4f ev = __builtin_nontemporal_load((const v4f*)(ep + (size_t)o * DIM_I + ib));

    float part = ev.x * (wv.x * xv.x)
               + ev.y * (wv.y * xv.y)
               + ev.z * (wv.z * xv.z)
               + ev.w * (wv.w * xv.w);

    // reduce across the 16-lane half (xor masks < 16 stay within the half)
    part += __shfl_xor(part, 1, 32);
    part += __shfl_xor(part, 2, 32);
    part += __shfl_xor(part, 4, 32);
    part += __shfl_xor(part, 8, 32);

    if (il == 0) {
      ob[(size_t)o * DIM_SP] += NOISE_F * part;   // unique (n,o) slot: safe RMW
    }
  }
}

extern "C" void kernel_launch(void* const* d_in, const int* in_sizes, int n_in,
                              void* d_out, int out_size, void* d_ws, size_t ws_size,
                              hipStream_t stream) {
  const float* x    = (const float*)d_in[0];   // [8, 64, 64, 16]
  const float* W    = (const float*)d_in[1];   // [128, 64]
  const float* bias = (const float*)d_in[2];   // [128]
  const float* eps  = (const float*)d_in[3];   // [8192, 128, 64]
  float* out = (float*)d_out;                  // [8, 128, 64, 16]
  (void)in_sizes; (void)n_in; (void)out_size; (void)d_ws; (void)ws_size;

  // Kernel 1: 4096 tile-waves, 8 waves (256 threads) per block -> 512 blocks.
  noiseconv_wmma_conv<<<512, 256, 0, stream>>>(x, W, bias, out);
  // Kernel 2: 8192 n-waves, 8 waves per block -> 1024 blocks.
  noiseconv_noisy_stream<<<1024, 256, 0, stream>>>(x, W, eps, out);
}